// MoELayer_44409961840863
// MI455X (gfx1250) — compile-verified
//
#include <hip/hip_runtime.h>
#include <math.h>

// ---------------------------------------------------------------------------
// MoE layer for MI455X (gfx1250, wave32, WMMA + async-to-LDS weight staging).
// B=1, S=512, D=256, H=2048, E=74, TOPK=6.
// Memory-bound on fp32 weight streaming (~310MB @ 23.3TB/s => ~13us floor):
// gather active tokens per expert, V_WMMA_F32_16X16X4_F32 for full-precision
// matmuls, GLOBAL_LOAD_ASYNC_TO_LDS_B128 double-buffered weight stream.
// ---------------------------------------------------------------------------

typedef __attribute__((ext_vector_type(2))) float v2f;
typedef __attribute__((ext_vector_type(8))) float v8f;
typedef __attribute__((ext_vector_type(4))) int   v4i;

#define Dm 256
#define Hm 2048
#define Sm 512
#define Em 74
#define TOPKm 6
#define MAX_TILES 32          // up to 512 tokens / 16 per expert
#define XP 260                // x-tile LDS pitch (260 % 64 == 4 -> conflict free)
#define HP 2052               // h-tile LDS pitch (2052 % 64 == 4 -> conflict free)
#define KC 32                 // K-chunk depth for async weight staging

#if defined(__AMDGCN__) && __has_builtin(__builtin_amdgcn_global_load_async_to_lds_b128)
#define USE_ASYNC_LDS 1
#else
#define USE_ASYNC_LDS 0
#endif

// ---------------------------------------------------------------------------
// Kernel 1: router. One block per token. logits -> softmax -> top6 -> gates.
// ---------------------------------------------------------------------------
__global__ void router_kernel(const float* __restrict__ x,
                              const float* __restrict__ Wr,
                              float* __restrict__ gates) {
  __shared__ float xs[Dm];
  __shared__ float probs[Em];
  const int s = blockIdx.x;
  const int t = threadIdx.x;

  for (int i = t; i < Dm; i += blockDim.x) xs[i] = x[(size_t)s * Dm + i];
  __syncthreads();

  if (t < Em) {
    float acc = 0.f;
    #pragma unroll 4
    for (int k = 0; k < Dm; ++k) acc = fmaf(xs[k], Wr[(size_t)k * Em + t], acc);
    probs[t] = acc;
  }
  __syncthreads();

  if (t == 0) {
    float mx = -1e30f;
    for (int e = 0; e < Em; ++e) mx = fmaxf(mx, probs[e]);
    float sum = 0.f;
    for (int e = 0; e < Em; ++e) { float p = __expf(probs[e] - mx); probs[e] = p; sum += p; }
    const float inv = 1.0f / sum;
    for (int e = 0; e < Em; ++e) probs[e] *= inv;
  }
  __syncthreads();

  for (int e = t; e < Em; e += blockDim.x) gates[(size_t)s * Em + e] = 0.f;
  __syncthreads();

  if (t == 0) {
    int   idxs[TOPKm];
    float vals[TOPKm];
    float ssum = 0.f;
    for (int j = 0; j < TOPKm; ++j) {
      float best = -1.f; int bi = 0;
      for (int e = 0; e < Em; ++e)
        if (probs[e] > best) { best = probs[e]; bi = e; }
      idxs[j] = bi; vals[j] = best; ssum += best;
      probs[bi] = -1.f;
    }
    const float inv = 1.0f / ssum;
    for (int j = 0; j < TOPKm; ++j)
      gates[(size_t)s * Em + idxs[j]] = vals[j] * inv;
  }
}

// ---------------------------------------------------------------------------
// Kernel 2: build compact per-expert token lists (deterministic serial scan).
// ---------------------------------------------------------------------------
__global__ void build_lists_kernel(const float* __restrict__ gates,
                                   int* __restrict__ nTok,
                                   int* __restrict__ tokList) {
  const int e = blockIdx.x;
  if (threadIdx.x == 0) {
    int c = 0;
    for (int s = 0; s < Sm; ++s)
      if (gates[(size_t)s * Em + e] > 0.f) tokList[(size_t)e * Sm + (c++)] = s;
    nTok[e] = c;
  }
}

// ---------------------------------------------------------------------------
// Kernel 3: load-balancing loss -> d_out[S*D].
// ---------------------------------------------------------------------------
__global__ void loss_kernel(const float* __restrict__ gates,
                            float* __restrict__ out) {
  __shared__ float counts[Em];
  const int t = threadIdx.x;
  if (t < Em) {
    float c = 0.f;
    for (int s = 0; s < Sm; ++s) c += gates[(size_t)s * Em + t];
    counts[t] = c;
  }
  __syncthreads();
  if (t == 0) {
    float tot = 0.f;
    for (int e = 0; e < Em; ++e) tot += counts[e];
    const float scale = (float)Em / tot;
    float l = 0.f;
    for (int e = 0; e < Em; ++e) { float d = counts[e] * scale - 1.f; l += d * d; }
    out[(size_t)Sm * Dm] = l / (float)Em;
  }
}

// ---------------------------------------------------------------------------
// Kernel 4: zero the output image (scatter-add target).
// ---------------------------------------------------------------------------
__global__ void zero_kernel(float* __restrict__ out, int n) {
  int i = blockIdx.x * blockDim.x + threadIdx.x;
  if (i < n) out[i] = 0.f;
}

// ---------------------------------------------------------------------------
// Async weight staging: copy a [KC x 16] fp32 tile (row stride ldw floats)
// into a per-wave LDS buffer.  4x GLOBAL_LOAD_ASYNC_TO_LDS_B128 per wave
// (16B/lane, bypasses VGPRs, tracked by ASYNCcnt).  Fallback: float4 copy.
// ---------------------------------------------------------------------------
__device__ __forceinline__ void stage_chunk(const float* __restrict__ gsrc,
                                            size_t ldw, float* lbuf, int lane) {
  #pragma unroll
  for (int j = 0; j < (KC * 16) / (32 * 4); ++j) {   // 4 instructions
    const int idx = (j * 32 + lane) * 4;             // float index in tile
    const int row = idx >> 4;                        // / 16 cols
    const int col = idx & 15;
#if USE_ASYNC_LDS
    __builtin_amdgcn_global_load_async_to_lds_b128(
        (__attribute__((address_space(1))) v4i*)(gsrc + (size_t)row * ldw + col),
        (__attribute__((address_space(3))) v4i*)(lbuf + idx),
        0, 0);
#else
    const float4 v = *(const float4*)(gsrc + (size_t)row * ldw + col);
    *(float4*)(lbuf + idx) = v;
#endif
  }
}

__device__ __forceinline__ void wait_async_le4() {
#if USE_ASYNC_LDS
  asm volatile("s_wait_asynccnt 0x4" ::: "memory");
#endif
}
__device__ __forceinline__ void wait_async_0() {
#if USE_ASYNC_LDS
  asm volatile("s_wait_asynccnt 0x0" ::: "memory");
#endif
}

// ---------------------------------------------------------------------------
// Kernel 5: grouped expert FFN with V_WMMA_F32_16X16X4_F32.
// grid.x = expert * MAX_TILES + tile ; 256 threads = 8 wave32.
// Per tile: gather 16 token rows of x -> LDS ; GEMM1 (16x256x2048) + GELU
// -> h tile in LDS ; GEMM2 (16x2048x256) ; bias ; gate-scale ; atomic scatter.
// Weights stream through per-wave double-buffered LDS chunks (async DMA).
// WMMA fragment mapping per ISA 7.12.2: lane&15 = M (A) / N (B,C,D);
// lane>>4 selects K-half (A/B: K = 2*half + vgpr) and M-half (C/D: M = 8*half + vgpr).
// EXEC is all-ones throughout the WMMA loops (uniform control flow).
// ---------------------------------------------------------------------------
__global__ __launch_bounds__(256) void
ffn_kernel(const float* __restrict__ x,
           const float* __restrict__ W1, const float* __restrict__ b1,
           const float* __restrict__ W2, const float* __restrict__ b2,
           const float* __restrict__ gates,
           const int* __restrict__ nTok, const int* __restrict__ tokList,
           float* __restrict__ out) {
  extern __shared__ float smem[];
  float* xs = smem;                      // [16][XP]
  float* hs = smem + 16 * XP;            // [16][HP]
  float* wb = hs + 16 * HP;              // 8 waves x 2 x [KC][16] weight buffers
  __shared__ int   s_tok[16];
  __shared__ float s_gate[16];

  const int e    = blockIdx.x / MAX_TILES;
  const int tile = blockIdx.x % MAX_TILES;
  const int ne   = nTok[e];
  if (tile * 16 >= ne) return;
  const int rows = min(16, ne - tile * 16);
  const int* tl  = tokList + (size_t)e * Sm + tile * 16;

  const int lane = threadIdx.x & 31;
  const int wave = threadIdx.x >> 5;
  const int l16  = lane & 15;        // A-row M / B-col N / C-col N
  const int half = lane >> 4;        // lane half
  const int kOfs = half * 2;         // K base for this half in A/B frags
  const int mRowBase = half * 8;     // C/D row base for this half

  float* buf0 = wb + wave * (2 * KC * 16);
  float* buf1 = buf0 + KC * 16;

  if (threadIdx.x < 16) {
    const int m = threadIdx.x;
    if (m < rows) {
      const int t = tl[m];
      s_tok[m]  = t;
      s_gate[m] = gates[(size_t)t * Em + e];
    } else {
      s_tok[m]  = 0;
      s_gate[m] = 0.f;
    }
  }
  // Gather 16 token rows of x into LDS (zero-pad missing rows).
  for (int i = threadIdx.x; i < 16 * Dm; i += blockDim.x) {
    const int m = i >> 8;            // Dm == 256
    const int k = i & (Dm - 1);
    float v = 0.f;
    if (m < rows) v = x[(size_t)tl[m] * Dm + k];
    xs[m * XP + k] = v;
  }
  __syncthreads();

  const float* W1e = W1 + (size_t)e * Dm * Hm;
  const float* b1e = b1 + (size_t)e * Hm;

  // ---- GEMM1: h = GELU(x_tile * W1_e + b1_e), N-chunks of 16 over H ----
  for (int nc = wave; nc < Hm / 16; nc += 8) {
    const int hbase = nc * 16;
    const float* gW = W1e + hbase;           // column window [hbase, hbase+16)
    v8f acc = {};
    stage_chunk(gW, Hm, buf0, lane);         // prologue: chunk 0
    constexpr int NCH = Dm / KC;             // 8 chunks
    for (int kc = 0; kc < NCH; ++kc) {
      float* cur = (kc & 1) ? buf1 : buf0;
      float* nxt = (kc & 1) ? buf0 : buf1;
      if (kc + 1 < NCH) {
        stage_chunk(gW + (size_t)(kc + 1) * KC * Hm, Hm, nxt, lane);
        wait_async_le4();                    // oldest 4 (cur chunk) complete
      } else {
        wait_async_0();
      }
      const int kbase = kc * KC;
      #pragma unroll
      for (int kk = 0; kk < KC; kk += 4) {
        v2f a, b;
        a.x = xs[l16 * XP + kbase + kk + kOfs];
        a.y = xs[l16 * XP + kbase + kk + kOfs + 1];
        b.x = cur[(kk + kOfs    ) * 16 + l16];
        b.y = cur[(kk + kOfs + 1) * 16 + l16];
        acc = __builtin_amdgcn_wmma_f32_16x16x4_f32(false, a, false, b,
                                                    (short)0, acc, false, false);
      }
    }
    const float bv = b1e[hbase + l16];
    #pragma unroll
    for (int r = 0; r < 8; ++r) {
      float v = acc[r] + bv;
      v = 0.5f * v * (1.0f + erff(v * 0.70710678118654752f));  // exact GELU
      hs[(mRowBase + r) * HP + hbase + l16] = v;
    }
  }
  __syncthreads();

  const float* W2e = W2 + (size_t)e * Hm * Dm;
  const float* b2e = b2 + (size_t)e * Dm;

  // ---- GEMM2: y = h * W2_e + b2_e, N-chunks of 16 over D ----
  for (int nc = wave; nc < Dm / 16; nc += 8) {
    const int dbase = nc * 16;
    const float* gW = W2e + dbase;
    v8f acc = {};
    stage_chunk(gW, Dm, buf0, lane);
    constexpr int NCH = Hm / KC;             // 64 chunks
    for (int kc = 0; kc < NCH; ++kc) {
      float* cur = (kc & 1) ? buf1 : buf0;
      float* nxt = (kc & 1) ? buf0 : buf1;
      if (kc + 1 < NCH) {
        stage_chunk(gW + (size_t)(kc + 1) * KC * Dm, Dm, nxt, lane);
        wait_async_le4();
      } else {
        wait_async_0();
      }
      const int kbase = kc * KC;
      #pragma unroll
      for (int kk = 0; kk < KC; kk += 4) {
        v2f a, b;
        a.x = hs[l16 * HP + kbase + kk + kOfs];
        a.y = hs[l16 * HP + kbase + kk + kOfs + 1];
        b.x = cur[(kk + kOfs    ) * 16 + l16];
        b.y = cur[(kk + kOfs + 1) * 16 + l16];
        acc = __builtin_amdgcn_wmma_f32_16x16x4_f32(false, a, false, b,
                                                    (short)0, acc, false, false);
      }
    }
    const float bv2 = b2e[dbase + l16];
    #pragma unroll
    for (int r = 0; r < 8; ++r) {
      const int m = mRowBase + r;
      if (m < rows) {
        const float v = (acc[r] + bv2) * s_gate[m];
        atomicAdd(&out[(size_t)s_tok[m] * Dm + dbase + l16], v);
      }
    }
  }
}

// ---------------------------------------------------------------------------
// Host-side launcher.
// ---------------------------------------------------------------------------
extern "C" void kernel_launch(void* const* d_in, const int* in_sizes, int n_in,
                              void* d_out, int out_size, void* d_ws, size_t ws_size,
                              hipStream_t stream) {
  const float* x  = (const float*)d_in[0];   // [1,512,256]
  const float* Wr = (const float*)d_in[1];   // [256,74]
  const float* W1 = (const float*)d_in[2];   // [74,256,2048]
  const float* b1 = (const float*)d_in[3];   // [74,2048]
  const float* W2 = (const float*)d_in[4];   // [74,2048,256]
  const float* b2 = (const float*)d_in[5];   // [74,256]
  float* out = (float*)d_out;                // [512*256] out + [1] loss

  // Workspace layout (fp32-aligned):
  //   gates  : S*E floats   @ 0
  //   nTok   : E ints       @ gatesBytes
  //   tokList: E*S ints     @ gatesBytes + 512
  char* ws = (char*)d_ws;
  const size_t gatesBytes = (size_t)Sm * Em * sizeof(float);   // 151552
  float* gates   = (float*)(ws);
  int*   nTok    = (int*)(ws + gatesBytes);
  int*   tokList = (int*)(ws + gatesBytes + 512);

  // 1. zero output image
  {
    const int n = Sm * Dm;
    zero_kernel<<<(n + 255) / 256, 256, 0, stream>>>(out, n);
  }
  // 2. router -> dense gates
  router_kernel<<<Sm, 128, 0, stream>>>(x, Wr, gates);
  // 3. per-expert token lists
  build_lists_kernel<<<Em, 32, 0, stream>>>(gates, nTok, tokList);
  // 4. load-balancing loss
  loss_kernel<<<1, 128, 0, stream>>>(gates, out);
  // 5. grouped expert FFN (WMMA + async weight staging)
  {
    const size_t smemBytes =
        (size_t)(16 * XP + 16 * HP + 8 * 2 * KC * 16) * sizeof(float); // ~177KB
    dim3 grid(Em * MAX_TILES);
    ffn_kernel<<<grid, 256, smemBytes, stream>>>(x, W1, b1, W2, b2,
                                                 gates, nTok, tokList, out);
  }
}